// TOGL_3470333575807
// MI455X (gfx1250) — compile-verified
//
#include <hip/hip_runtime.h>

#define N_NODES 10000
#define N_EDGES 160000
#define N_FEAT  128
#define HIDDEN  256
#define N_FILT  4
#define KEY_RANGE (N_NODES + 1)   // key == N_NODES means "inactive edge"

typedef float v2f __attribute__((ext_vector_type(2)));
typedef float v8f __attribute__((ext_vector_type(8)));

// ---------------------------------------------------------------------------
// GEMM1: H = relu(X @ W1 + b1), (10000x128)@(128x256), exact f32 via
// V_WMMA_F32_16X16X4_F32. One wave per 16x16 output tile; 10000 tiles total.
// ---------------------------------------------------------------------------
__global__ void togl_gemm1_wmma(const float* __restrict__ X,
                                const float* __restrict__ W1,
                                const float* __restrict__ b1,
                                float* __restrict__ H) {
  int wave = (blockIdx.x * blockDim.x + threadIdx.x) >> 5;  // wave-uniform
  int lane = threadIdx.x & 31;
  int tileN = wave & 15;            // HIDDEN/16 = 16 tiles in N
  int tileM = wave >> 4;            // N_NODES/16 = 625 tiles in M
  if (tileM >= N_NODES / 16) return;  // wave-uniform exit (EXEC all-1 for WMMA)

  int half = lane >> 4;             // 0: K pair {0,1}; 1: K pair {2,3}
  int l16  = lane & 15;
  int arow = tileM * 16 + l16;      // A: M = lane&15
  int bcol = tileN * 16 + l16;      // B/C: N = lane&15

  v8f acc = {};
  #pragma unroll 4
  for (int k0 = 0; k0 < N_FEAT; k0 += 4) {
    int ka = k0 + half * 2;
    v2f a, b;
    a.x = X[arow * N_FEAT + ka];          // contiguous pair -> b64 load
    a.y = X[arow * N_FEAT + ka + 1];
    b.x = W1[ka * HIDDEN + bcol];
    b.y = W1[(ka + 1) * HIDDEN + bcol];
    acc = __builtin_amdgcn_wmma_f32_16x16x4_f32(
        /*neg_a=*/false, a, /*neg_b=*/false, b,
        /*c_mod=*/(short)0, acc, /*reuse_a=*/false, /*reuse_b=*/false);
  }

  float bias = b1[bcol];
  #pragma unroll
  for (int r = 0; r < 8; ++r) {
    int m = tileM * 16 + half * 8 + r;    // C layout: VGPR r -> M = r (+8 hi half)
    float v = acc[r] + bias;
    H[m * HIDDEN + bcol] = v > 0.0f ? v : 0.0f;
  }
}

// ---------------------------------------------------------------------------
// GEMM2: Ft[filt][node] = H[node,:] . W2[:,filt] + b2[filt]   (N=4 -> VALU)
// ---------------------------------------------------------------------------
__global__ void togl_gemm2(const float* __restrict__ H,
                           const float* __restrict__ W2,
                           const float* __restrict__ b2,
                           float* __restrict__ Ft) {
  int t = blockIdx.x * blockDim.x + threadIdx.x;
  if (t >= N_NODES * N_FILT) return;
  int node = t >> 2;
  int filt = t & 3;
  float acc = b2[filt];
  const float* h = H + node * HIDDEN;
  #pragma unroll 4
  for (int k = 0; k < HIDDEN; ++k) acc = fmaf(h[k], W2[k * N_FILT + filt], acc);
  Ft[filt * N_NODES + node] = acc;
}

// ---------------------------------------------------------------------------
// Stable rank == pos from jnp.argsort: pos[v] = #{u: f[u]<f[v] or (==, u<v)}
// O(n^2) but the 40KB filtration vector is L2-resident and broadcast-read.
// ---------------------------------------------------------------------------
__global__ void togl_rank(const float* __restrict__ Ft, int* __restrict__ pos) {
  int filt = blockIdx.y;
  int node = blockIdx.x * blockDim.x + threadIdx.x;
  if (node >= N_NODES) return;
  const float* f = Ft + filt * N_NODES;
  float fv = f[node];
  int cnt = 0;
  for (int j = 0; j < N_NODES; ++j) {
    float fj = f[j];
    cnt += (fj < fv) || ((fj == fv) && (j < node));
  }
  pos[filt * N_NODES + node] = cnt;
}

__global__ void togl_zero_i32(int* __restrict__ p, int n) {
  int t = blockIdx.x * blockDim.x + threadIdx.x;
  if (t < n) p[t] = 0;
}

// Counting-sort pass 1: histogram of edge keys (key = active ? pos[ev] : N).
__global__ void togl_edge_hist(const int* __restrict__ eu, const int* __restrict__ ev,
                               const int* __restrict__ pos, int* __restrict__ hist) {
  int filt = blockIdx.y;
  int e = blockIdx.x * blockDim.x + threadIdx.x;
  if (e >= N_EDGES) return;
  const int* p = pos + filt * N_NODES;
  int pu = p[eu[e] - 1];
  int pv = p[ev[e] - 1];
  int key = (pu < pv) ? pv : N_NODES;
  atomicAdd(&hist[filt * KEY_RANGE + key], 1);
}

// Exclusive scan of 10001 buckets, one lane per filtration (trivial work).
__global__ void togl_scan(const int* __restrict__ hist, int* __restrict__ cur) {
  int filt = threadIdx.x;
  if (filt >= N_FILT) return;
  int acc = 0;
  for (int k = 0; k < KEY_RANGE; ++k) {
    cur[filt * KEY_RANGE + k] = acc;
    acc += hist[filt * KEY_RANGE + k];
  }
}

// Counting-sort pass 2: scatter (within-bucket order is output-invariant).
__global__ void togl_edge_scatter(const int* __restrict__ eu, const int* __restrict__ ev,
                                  const int* __restrict__ pos, int* __restrict__ cur,
                                  int* __restrict__ su, int* __restrict__ sv,
                                  int* __restrict__ sk) {
  int filt = blockIdx.y;
  int e = blockIdx.x * blockDim.x + threadIdx.x;
  if (e >= N_EDGES) return;
  const int* p = pos + filt * N_NODES;
  int u = eu[e] - 1;
  int v = ev[e] - 1;
  int pu = p[u];
  int pv = p[v];
  int key = (pu < pv) ? pv : N_NODES;
  int idx = atomicAdd(&cur[filt * KEY_RANGE + key], 1);
  size_t o = (size_t)filt * N_EDGES + idx;
  su[o] = u; sv[o] = v; sk[o] = key;
}

// ---------------------------------------------------------------------------
// Serial Kruskal/union-find per filtration. One workgroup per filtration on
// its own WGP; parent/rank/minlab/death (160 KB) live in LDS so the dependent
// pointer chases are LDS-latency, not L2-latency. 4 filtrations run
// concurrently on 4 WGPs.
// ---------------------------------------------------------------------------
extern __shared__ int togl_smem[];
__global__ void togl_union_find(const int* __restrict__ su, const int* __restrict__ sv,
                                const int* __restrict__ sk, float* __restrict__ out) {
  int filt = blockIdx.x;
  int* parent = togl_smem;
  int* rnk    = togl_smem + N_NODES;
  int* minlab = togl_smem + 2 * N_NODES;
  float* death = (float*)(togl_smem + 3 * N_NODES);

  for (int i = threadIdx.x; i < N_NODES; i += blockDim.x) {
    parent[i] = i; rnk[i] = 0; minlab[i] = i;
    death[i] = __builtin_huge_valf();
  }
  __syncthreads();

  if (threadIdx.x == 0) {
    const int* u_ = su + (size_t)filt * N_EDGES;
    const int* v_ = sv + (size_t)filt * N_EDGES;
    const int* k_ = sk + (size_t)filt * N_EDGES;
    for (int e = 0; e < N_EDGES; ++e) {
      int key = k_[e];
      if (key >= N_NODES) break;   // sorted: all inactive edges are at the end
      int u = u_[e];
      int v = v_[e];
      int ru = u; { int p = parent[ru]; while (p != ru) { ru = p; p = parent[ru]; } }
      int rv = v; { int p = parent[rv]; while (p != rv) { rv = p; p = parent[rv]; } }
      parent[u] = ru;              // path compression (as in reference)
      parent[v] = rv;
      if (ru != rv) {
        int mu = minlab[ru], mv = minlab[rv];
        int loser  = mu > mv ? mu : mv;
        int winner = mu > mv ? mv : mu;
        int rru = rnk[ru], rrv = rnk[rv];
        int big   = (rru >= rrv) ? ru : rv;
        int small = (rru >= rrv) ? rv : ru;
        parent[small] = big;
        if (rru == rrv) rnk[big] = rru + 1;
        minlab[big] = winner;
        death[loser] = (float)(key - 1);   // death = merge_step - 1
      }
    }
  }
  __syncthreads();

  for (int i = threadIdx.x; i < N_NODES; i += blockDim.x) {
    size_t o = ((size_t)filt * N_NODES + i) * 2;
    out[o + 0] = 0.0f;       // birth
    out[o + 1] = death[i];   // death (+inf for survivors)
  }
}

// ---------------------------------------------------------------------------
extern "C" void kernel_launch(void* const* d_in, const int* in_sizes, int n_in,
                              void* d_out, int out_size, void* d_ws, size_t ws_size,
                              hipStream_t stream) {
  const float* X  = (const float*)d_in[0];
  const float* W1 = (const float*)d_in[1];
  const float* b1 = (const float*)d_in[2];
  const float* W2 = (const float*)d_in[3];
  const float* b2 = (const float*)d_in[4];
  const int* edge = (const int*)d_in[5];
  const int* eu = edge;
  const int* ev = edge + N_EDGES;

  char* ws = (char*)d_ws;
  size_t off = 0;
  auto alloc = [&](size_t bytes) -> void* {
    void* p = ws + off;
    off = (off + bytes + 255) & ~(size_t)255;
    return p;
  };
  float* H    = (float*)alloc((size_t)N_NODES * HIDDEN * 4);      // 10.24 MB
  float* Ft   = (float*)alloc((size_t)N_FILT * N_NODES * 4);
  int*   pos  = (int*)  alloc((size_t)N_FILT * N_NODES * 4);
  int*   hist = (int*)  alloc((size_t)N_FILT * KEY_RANGE * 4);
  int*   cur  = (int*)  alloc((size_t)N_FILT * KEY_RANGE * 4);
  int*   su   = (int*)  alloc((size_t)N_FILT * N_EDGES * 4);
  int*   sv   = (int*)  alloc((size_t)N_FILT * N_EDGES * 4);
  int*   sk   = (int*)  alloc((size_t)N_FILT * N_EDGES * 4);      // total ~18.56 MB

  // 1) MLP
  togl_gemm1_wmma<<<(N_NODES / 16) * (HIDDEN / 16) * 32 / 256, 256, 0, stream>>>(X, W1, b1, H);
  togl_gemm2<<<(N_NODES * N_FILT + 255) / 256, 256, 0, stream>>>(H, W2, b2, Ft);

  // 2) stable ranks (argsort-equivalent pos)
  togl_rank<<<dim3((N_NODES + 255) / 256, N_FILT), 256, 0, stream>>>(Ft, pos);

  // 3) counting sort of edges by activation step
  togl_zero_i32<<<(N_FILT * KEY_RANGE + 255) / 256, 256, 0, stream>>>(hist, N_FILT * KEY_RANGE);
  togl_edge_hist<<<dim3((N_EDGES + 255) / 256, N_FILT), 256, 0, stream>>>(eu, ev, pos, hist);
  togl_scan<<<1, 32, 0, stream>>>(hist, cur);
  togl_edge_scatter<<<dim3((N_EDGES + 255) / 256, N_FILT), 256, 0, stream>>>(
      eu, ev, pos, cur, su, sv, sk);

  // 4) union-find persistence, LDS-resident state, one WGP per filtration
  togl_union_find<<<N_FILT, 64, 4 * N_NODES * 4, stream>>>(su, sv, sk, (float*)d_out);
}